// CommAwareGAT_53025666237104
// MI455X (gfx1250) — compile-verified
//
#include <hip/hip_runtime.h>
#include <math.h>

typedef float v2f __attribute__((ext_vector_type(2)));
typedef float v8f __attribute__((ext_vector_type(8)));

#define N_NODES 100000
#define N_EDGES 1250000
#define F_IN    128
#define F_OUT   64
#define N_TILES (N_NODES / 16)   // 6250, exact

// ---------------------------------------------------------------------------
// h = x @ W1 via V_WMMA_F32_16X16X4_F32.  One wave32 per 16-row tile, full 64
// output cols (4 x 16-wide col tiles), K=128 in steps of 4 -> 32x4 = 128 WMMA
// per tile.  8 waves per 256-thread block.
// A 16x4 layout: lane(m=lane&15); VGPR0 = K(0|2), VGPR1 = K(1|3) by lane half.
// B 4x16 layout: symmetric (row striped across lanes, K split by lane half).
// C 16x16: VGPR v -> M = v + 8*(lane>=16), N = lane&15 (+16*coltile).
// ---------------------------------------------------------------------------
__global__ __launch_bounds__(256)
void gat_gemm_wmma(const float* __restrict__ x, const float* __restrict__ W1,
                   float* __restrict__ h) {
  const int lane  = threadIdx.x & 31;
  const int wave  = threadIdx.x >> 5;
  const int tile  = blockIdx.x * 8 + wave;
  if (tile >= N_TILES) return;           // wave-uniform -> EXEC stays all-1s

  const int m0    = tile * 16;
  const int mlane = lane & 15;
  const int khigh = lane >> 4;           // 0: K=0,1  1: K=2,3
  const int nlane = lane & 15;

  const float* xrow = x + (size_t)(m0 + mlane) * F_IN;

  v8f c0 = {}, c1 = {}, c2 = {}, c3 = {};

  #pragma unroll 4
  for (int k = 0; k < F_IN; k += 4) {
    const int ka = k + 2 * khigh;
    v2f a;
    a.x = xrow[ka];
    a.y = xrow[ka + 1];

    const float* wr0 = W1 + (size_t)ka * F_OUT;        // row K=ka
    const float* wr1 = wr0 + F_OUT;                    // row K=ka+1
    v2f b0, b1, b2, b3;
    b0.x = wr0[nlane];      b0.y = wr1[nlane];
    b1.x = wr0[16 + nlane]; b1.y = wr1[16 + nlane];
    b2.x = wr0[32 + nlane]; b2.y = wr1[32 + nlane];
    b3.x = wr0[48 + nlane]; b3.y = wr1[48 + nlane];

    c0 = __builtin_amdgcn_wmma_f32_16x16x4_f32(false, a, false, b0, (short)0, c0, false, false);
    c1 = __builtin_amdgcn_wmma_f32_16x16x4_f32(false, a, false, b1, (short)0, c1, false, false);
    c2 = __builtin_amdgcn_wmma_f32_16x16x4_f32(false, a, false, b2, (short)0, c2, false, false);
    c3 = __builtin_amdgcn_wmma_f32_16x16x4_f32(false, a, false, b3, (short)0, c3, false, false);
  }

  #pragma unroll
  for (int v = 0; v < 8; ++v) {
    const int row = m0 + v + 8 * khigh;
    float* hrow = h + (size_t)row * F_OUT + nlane;
    hrow[0]  = c0[v];
    hrow[16] = c1[v];
    hrow[32] = c2[v];
    hrow[48] = c3[v];
  }
}

// ---------------------------------------------------------------------------
// Per-node attention coefficients: ai = h . Wp[0:64], aj = h . Wp[64:128].
// Factorizes the per-edge 128-wide dot into two scalar gathers.
// ---------------------------------------------------------------------------
__global__ __launch_bounds__(256)
void gat_coef(const float* __restrict__ h, const float* __restrict__ Wp,
              float* __restrict__ ai, float* __restrict__ aj) {
  const int n = blockIdx.x * blockDim.x + threadIdx.x;
  if (n >= N_NODES) return;
  const float* hr = h + (size_t)n * F_OUT;
  float si = 0.f, sj = 0.f;
  #pragma unroll 8
  for (int f = 0; f < F_OUT; ++f) {
    const float v = hr[f];
    si += v * Wp[f];
    sj += v * Wp[F_OUT + f];
  }
  ai[n] = si;
  aj[n] = sj;
}

// out <- bias broadcast (atomics then accumulate the segment sum), denom <- 0
__global__ __launch_bounds__(256)
void gat_init(float* __restrict__ out, float* __restrict__ denom,
              const float* __restrict__ bias) {
  const int i = blockIdx.x * blockDim.x + threadIdx.x;
  if (i < N_NODES * F_OUT) out[i] = bias[i & (F_OUT - 1)];
  if (i < N_NODES) denom[i] = 0.f;
}

// ---------------------------------------------------------------------------
// Per-edge score: s = ai[dst] + aj[src] + bp; leaky-relu(0.2); num = exp(s);
// denom[dst] += num  (hardware f32 atomic, relaxed, device scope).
// ---------------------------------------------------------------------------
__global__ __launch_bounds__(256)
void gat_edge(const int* __restrict__ src, const int* __restrict__ dst,
              const float* __restrict__ ai, const float* __restrict__ aj,
              const float* __restrict__ bp, float* __restrict__ num,
              float* __restrict__ denom) {
  const int e = blockIdx.x * blockDim.x + threadIdx.x;
  if (e >= N_EDGES) return;
  const int d = dst[e];
  const int s = src[e];
  float sc = ai[d] + aj[s] + bp[0];
  sc = (sc >= 0.f) ? sc : 0.2f * sc;
  const float nu = __expf(sc);
  num[e] = nu;
  __hip_atomic_fetch_add(&denom[d], nu, __ATOMIC_RELAXED, __HIP_MEMORY_SCOPE_AGENT);
}

// ---------------------------------------------------------------------------
// out[dst] += h[src] * alpha,  alpha = num / (denom[src] + 1e-16)
// NOTE: reference gathers denominator at SRC (its quirk) -> replicate exactly.
// One wave per edge: lane handles 2 features (float2 load + 2 f32 atomics);
// src[e]/num[e]/denom loads are wave-uniform.
// ---------------------------------------------------------------------------
__global__ __launch_bounds__(256)
void gat_aggregate(const int* __restrict__ src, const int* __restrict__ dst,
                   const float* __restrict__ h, const float* __restrict__ num,
                   const float* __restrict__ denom, float* __restrict__ out) {
  const int gid = blockIdx.x * blockDim.x + threadIdx.x;
  if (gid >= N_EDGES * 32) return;
  const int e  = gid >> 5;
  const int fp = (gid & 31) * 2;
  const int s  = src[e];
  const int d  = dst[e];
  const float alpha = num[e] / (denom[s] + 1e-16f);
  const float2 hj = *(const float2*)(h + (size_t)s * F_OUT + fp);
  float* o = out + (size_t)d * F_OUT + fp;
  __hip_atomic_fetch_add(&o[0], hj.x * alpha, __ATOMIC_RELAXED, __HIP_MEMORY_SCOPE_AGENT);
  __hip_atomic_fetch_add(&o[1], hj.y * alpha, __ATOMIC_RELAXED, __HIP_MEMORY_SCOPE_AGENT);
}

extern "C" void kernel_launch(void* const* d_in, const int* in_sizes, int n_in,
                              void* d_out, int out_size, void* d_ws, size_t ws_size,
                              hipStream_t stream) {
  const float* x    = (const float*)d_in[0];  // (1, N, 128)
  const int*   ei   = (const int*)  d_in[1];  // (1, 2, E): [0]=src, [1]=dst
  /* d_in[2] rank_mapping: unused by reference */
  const float* W1   = (const float*)d_in[3];  // (128, 64)
  const float* Wp   = (const float*)d_in[4];  // (128, 1)
  const float* bp   = (const float*)d_in[5];  // (1,)
  const float* bias = (const float*)d_in[6];  // (64,)
  float* out = (float*)d_out;                 // (1, N, 64)

  const int* src = ei;             // edge_index[:,0,:]
  const int* dst = ei + N_EDGES;   // edge_index[:,1,:]

  // Workspace layout (floats): h | ai | aj | denom | num  (~31.8 MB total)
  float* ws    = (float*)d_ws;
  float* h     = ws;
  float* ai    = h  + (size_t)N_NODES * F_OUT;
  float* aj    = ai + N_NODES;
  float* denom = aj + N_NODES;
  float* num   = denom + N_NODES;

  gat_init     <<<(N_NODES * F_OUT + 255) / 256, 256, 0, stream>>>(out, denom, bias);
  gat_gemm_wmma<<<(N_TILES + 7) / 8,             256, 0, stream>>>(x, W1, h);
  gat_coef     <<<(N_NODES + 255) / 256,         256, 0, stream>>>(h, Wp, ai, aj);
  gat_edge     <<<(N_EDGES + 255) / 256,         256, 0, stream>>>(src, dst, ai, aj, bp, num, denom);
  gat_aggregate<<<(N_EDGES * 32 + 255) / 256,    256, 0, stream>>>(src, dst, h, num, denom, out);
}